// ReconsLoss_441
// MI455X (gfx1250) — compile-verified
//
#include <hip/hip_runtime.h>
#include <hip/hip_bf16.h>
#include <math.h>

#define NUM   64
#define N     256
#define SPEC  1024
#define COLS  1280   // N + SPEC

typedef __attribute__((ext_vector_type(2))) float v2f;
typedef __attribute__((ext_vector_type(8))) float v8f;

// ---------------------------------------------------------------------------
// Kernel 1: per-batch in-place Gauss-Jordan inversion, fully LDS-resident.
// One 1024-thread block (32 waves) per batch. The 256x256 f32 matrix (256 KB)
// lives in dynamic LDS -- CDNA5's 320 KB/WGP makes this possible.
// M = XS_Int_pred[b] * S_N is formed during the LDS fill (no staging kernel).
// In-place GJ (no augmentation):
//   d = 1/A[k][k];  row_k *= d;  A[k][k] = d
//   for r != k: f = A[r][k];  row_r -= f*row_k (cols != k);  A[r][k] = -f*d
// ---------------------------------------------------------------------------
__global__ void invert_kernel(const float* __restrict__ XS,
                              const float* __restrict__ S_N,
                              float* __restrict__ inv) {
    extern __shared__ float A[];            // N*N floats = 256 KB
    __shared__ float rowk[N];
    __shared__ float fac[N];
    __shared__ float dinv_s;

    const int b = blockIdx.x;
    const int t = threadIdx.x;              // 0..1023
    const float* XSb = XS + (size_t)b * N * N;

    // fill LDS with M = XS[b] * S_N  (64 elements per thread, coalesced)
    for (int idx = t; idx < N * N; idx += 1024)
        A[idx] = XSb[idx] * S_N[idx];
    __syncthreads();

    const int c  = t & (N - 1);             // column owned in elimination
    const int r0 = t >> 8;                  // 0..3: row phase

    for (int k = 0; k < N; ++k) {
        if (t == 0) dinv_s = 1.0f / A[k * N + k];
        __syncthreads();
        const float d = dinv_s;

        // scale pivot row (incl. diagonal -> 1), cache row + factors
        if (t < N) {
            const float e = A[k * N + t] * d;
            A[k * N + t] = e;
            rowk[t] = e;                    // rowk[k] == 1
            fac[t]  = (t == k) ? 0.0f : A[t * N + k];
        }
        __syncthreads();

        // rank-1 elimination on LDS: thread t owns column c, rows r0+4*i
        const float rc = rowk[c];
        #pragma unroll 4
        for (int r = r0; r < N; r += 4) {
            A[r * N + c] -= fac[r] * rc;    // fac[k]==0 -> row k untouched
        }
        __syncthreads();

        // fix up column k and the pivot element
        if (t < N) {
            A[t * N + k] = (t == k) ? d : -fac[t] * d;
        }
        __syncthreads();
    }

    // write the inverse out (dense 256x256)
    float* out = inv + (size_t)b * N * N;
    for (int idx = t; idx < N * N; idx += 1024)
        out[idx] = A[idx];
}

// ---------------------------------------------------------------------------
// Kernel 2: re_1[b] = inv(M[b]) (256x256) @ coded[b] (256x1280) in f32 WMMA.
// One wave per 16x64 output strip: 4 accumulators, A reused across 4 WMMAs.
// V_WMMA_F32_16X16X4_F32 layouts (ISA 7.12.2):
//   A 16x4: lane l (0..15) holds A[M=l, K=k0..k0+1] in v0,v1; lanes 16..31
//           hold K=k0+2..k0+3 for M=l-16.
//   B 4x16: v0 = row K=k0 (lanes 0-15, N=lane) / row K=k0+2 (lanes 16-31);
//           v1 = rows k0+1 / k0+3.
//   C 16x16: VGPR v, lanes 0-15 -> (M=v, N=lane); lanes 16-31 -> (M=v+8).
// ---------------------------------------------------------------------------
__global__ void gemm_wmma_kernel(const float* __restrict__ inv,
                                 const float* __restrict__ Bmat,
                                 float* __restrict__ re1) {
    const int lane = threadIdx.x & 31;
    const int wave = threadIdx.x >> 5;
    // 64 batches * 16 row-tiles * 20 col-strips (each strip = 64 cols)
    const int strip = blockIdx.x * (blockDim.x >> 5) + wave;
    const int b  = strip / (16 * 20);
    const int rm = strip % (16 * 20);
    const int rt = rm / 20;         // row tile   (16 rows)
    const int cs = rm % 20;         // col strip  (64 cols)

    const int half = lane >> 4;     // 0: lanes 0-15, 1: lanes 16-31
    const int lid  = lane & 15;

    const float* invA = inv + (size_t)b * N * N;
    const float* Bb   = Bmat + (size_t)b * N * COLS;
    const int arow  = rt * 16 + lid;
    const int cbase = cs * 64;

    v8f acc0 = {}, acc1 = {}, acc2 = {}, acc3 = {};

    for (int k0 = 0; k0 < N; k0 += 4) {
        const int ka = k0 + half * 2;
        v2f a;
        a.x = invA[(size_t)arow * N + ka];
        a.y = invA[(size_t)arow * N + ka + 1];

        const float* Br0 = Bb + (size_t)ka * COLS + cbase + lid;
        const float* Br1 = Br0 + COLS;

        v2f b0, b1, b2, b3;
        b0.x = Br0[0];  b0.y = Br1[0];
        b1.x = Br0[16]; b1.y = Br1[16];
        b2.x = Br0[32]; b2.y = Br1[32];
        b3.x = Br0[48]; b3.y = Br1[48];

        acc0 = __builtin_amdgcn_wmma_f32_16x16x4_f32(false, a, false, b0,
                                                     (short)0, acc0, false, false);
        acc1 = __builtin_amdgcn_wmma_f32_16x16x4_f32(false, a, false, b1,
                                                     (short)0, acc1, false, false);
        acc2 = __builtin_amdgcn_wmma_f32_16x16x4_f32(false, a, false, b2,
                                                     (short)0, acc2, false, false);
        acc3 = __builtin_amdgcn_wmma_f32_16x16x4_f32(false, a, false, b3,
                                                     (short)0, acc3, false, false);
    }

    // store: row = rt*16 + v + 8*half, col = cbase + 16*j + lid
    float* out = re1 + (size_t)b * N * COLS;
    #pragma unroll
    for (int v = 0; v < 8; ++v) {
        const int row = rt * 16 + v + 8 * half;
        float* orow = out + (size_t)row * COLS + cbase + lid;
        orow[0]  = acc0[v];
        orow[16] = acc1[v];
        orow[32] = acc2[v];
        orow[48] = acc3[v];
    }
}

// ---------------------------------------------------------------------------
// Kernel 3: per-row shifted gather + SNR.  f[j] = re1[b,i,(j+i)%1280], j<1024
// snr[b,i] = 10*log10( sum f^2 / sum (fgt-f)^2 ),  fgt = X_S[b,0]+X_S[b,1]
// ---------------------------------------------------------------------------
__global__ void snr_kernel(const float* __restrict__ re1,
                           const float* __restrict__ X_S,
                           float* __restrict__ snr) {
    const int row = blockIdx.x;          // 0 .. 64*256-1
    const int b = row >> 8;
    const int i = row & 255;
    const float* r1  = re1 + ((size_t)b * N + i) * COLS;
    const float* xs0 = X_S + (((size_t)b * 2 + 0) * N + i) * SPEC;
    const float* xs1 = X_S + (((size_t)b * 2 + 1) * N + i) * SPEC;
    const int t = threadIdx.x;           // 256 threads

    float up = 0.0f, lo = 0.0f;
    for (int j = t; j < SPEC; j += 256) {
        int c = j + i;
        if (c >= COLS) c -= COLS;        // (j+i) % 1280, j+i < 2304
        const float fj = r1[c];
        const float g  = xs0[j] + xs1[j];
        up += fj * fj;
        const float d = g - fj;
        lo += d * d;
    }

    __shared__ float su[256];
    __shared__ float sl[256];
    su[t] = up; sl[t] = lo;
    __syncthreads();
    for (int s = 128; s > 0; s >>= 1) {
        if (t < s) { su[t] += su[t + s]; sl[t] += sl[t + s]; }
        __syncthreads();
    }
    if (t == 0) snr[row] = 10.0f * log10f(su[0] / sl[0]);
}

// ---------------------------------------------------------------------------
// Kernel 4: mean over 16384 snr values -> out[0]=100-mean, out[1]=mean
// ---------------------------------------------------------------------------
__global__ void finalize_kernel(const float* __restrict__ snr,
                                float* __restrict__ out) {
    const int t = threadIdx.x;           // 256 threads
    float s = 0.0f;
    for (int j = t; j < NUM * N; j += 256) s += snr[j];
    __shared__ float sh[256];
    sh[t] = s;
    __syncthreads();
    for (int k = 128; k > 0; k >>= 1) {
        if (t < k) sh[t] += sh[t + k];
        __syncthreads();
    }
    if (t == 0) {
        const float mean = sh[0] / (float)(NUM * N);
        out[0] = 100.0f - mean;
        out[1] = mean;
    }
}

// ---------------------------------------------------------------------------
extern "C" void kernel_launch(void* const* d_in, const int* in_sizes, int n_in,
                              void* d_out, int out_size, void* d_ws, size_t ws_size,
                              hipStream_t stream) {
    const float* coded = (const float*)d_in[0];   // (64, 256, 1280)
    const float* XS    = (const float*)d_in[1];   // (64, 256, 256)
    const float* X_S   = (const float*)d_in[2];   // (64, 2, 256, 1024)
    const float* S_N   = (const float*)d_in[3];   // (256, 256)
    float* out = (float*)d_out;

    // workspace layout (floats)
    float* inv    = (float*)d_ws;                       // 64*256*256  = 16.8 MB
    float* re1    = inv + (size_t)NUM * N * N;          // 64*256*1280 = 83.9 MB
    float* snrbuf = re1 + (size_t)NUM * N * COLS;       // 64*256      = 64 KB

    // allow 256 KB dynamic LDS for the inversion kernel (CDNA5: 320 KB/WGP)
    const int ldsBytes = N * N * (int)sizeof(float);    // 262144
    (void)hipFuncSetAttribute((const void*)invert_kernel,
                              hipFuncAttributeMaxDynamicSharedMemorySize,
                              ldsBytes);

    // 1) LDS-resident Gauss-Jordan inversion, one 1024-thread block per batch
    invert_kernel<<<NUM, 1024, ldsBytes, stream>>>(XS, S_N, inv);
    // 2) f32 WMMA GEMM: 64 * 16 * 20 strips, 8 waves per 256-thread block
    {
        const int strips = NUM * 16 * 20;               // 20480
        gemm_wmma_kernel<<<strips / 8, 256, 0, stream>>>(inv, coded, re1);
    }
    // 3) per-row SNR
    snr_kernel<<<NUM * N, 256, 0, stream>>>(re1, X_S, snrbuf);
    // 4) reduce to the two outputs
    finalize_kernel<<<1, 256, 0, stream>>>(snrbuf, out);
}